// Model_88699664597656
// MI455X (gfx1250) — compile-verified
//
#include <hip/hip_runtime.h>
#include <math.h>

// ---------------- problem constants (match reference) ----------------
#define NUSERS 100000
#define NITEMS 50000
#define NNODES 150000           // NUSERS + NITEMS
#define H      128
#define NEDGES 1000000
#define NLAYERS 3
#define OUTW   512              // H * (NLAYERS + 1)
#define SLOPE  0.2f

typedef __attribute__((ext_vector_type(2))) float v2f;
typedef __attribute__((ext_vector_type(8))) float v8f;

// Native f32 atomic add (GLOBAL_ATOMIC_ADD_F32, no CAS loop).
__device__ __forceinline__ void atomAddF(float* p, float v) {
    (void)unsafeAtomicAdd(p, v);
}

// ---------------- 1) degree: deg[dst] += 1 for both directions ----------------
__global__ __launch_bounds__(256) void deg_kernel(const int* __restrict__ uidx,
                                                  const int* __restrict__ iidx,
                                                  float* __restrict__ deg, int nE) {
    int e = blockIdx.x * blockDim.x + threadIdx.x;
    if (e >= nE) return;
    atomAddF(&deg[uidx[e]], 1.0f);
    atomAddF(&deg[NUSERS + iidx[e]], 1.0f);
}

// ---------------- 2) dinv[n] = 1/sqrt(max(deg,1)) in place ----------------
__global__ __launch_bounds__(256) void dinv_kernel(float* __restrict__ deg, int n) {
    int i = blockIdx.x * blockDim.x + threadIdx.x;
    if (i >= n) return;
    deg[i] = 1.0f / sqrtf(fmaxf(deg[i], 1.0f));
}

// ---------------- 3) per-edge symmetric norm (same both directions) ----------------
__global__ __launch_bounds__(256) void enorm_kernel(const int* __restrict__ uidx,
                                                    const int* __restrict__ iidx,
                                                    const float* __restrict__ dinv,
                                                    float* __restrict__ nrm, int nE) {
    int e = blockIdx.x * blockDim.x + threadIdx.x;
    if (e >= nE) return;
    nrm[e] = dinv[uidx[e]] * dinv[NUSERS + iidx[e]];
}

// ---------------- 4) init: x0 = [user_embd; item_embd], out[:,0:128] = x0 ----------------
__global__ __launch_bounds__(256) void init_kernel(const float* __restrict__ ue,
                                                   const float* __restrict__ ie,
                                                   float* __restrict__ x0,
                                                   float* __restrict__ out) {
    int idx = blockIdx.x * blockDim.x + threadIdx.x;   // node*32 + c4
    if (idx >= NNODES * 32) return;
    int n = idx >> 5, c = idx & 31;
    const float4* src = (n < NUSERS) ? (const float4*)(ue + (size_t)n * H)
                                     : (const float4*)(ie + (size_t)(n - NUSERS) * H);
    float4 v = src[c];
    ((float4*)(x0 + (size_t)n * H))[c]    = v;
    ((float4*)(out + (size_t)n * OUTW))[c] = v;
}

// ---------------- 5) edge scatter: one wave per edge, both directions ----------------
// agg[item] += (x[user] + ef) * nrm ;  agg[user] += (x[item] + ef) * nrm
__global__ __launch_bounds__(256) void edge_kernel(const float* __restrict__ x,
                                                   const float* __restrict__ tt,
                                                   const float* __restrict__ nrm,
                                                   const int* __restrict__ uidx,
                                                   const int* __restrict__ iidx,
                                                   const int* __restrict__ tseq,
                                                   float* __restrict__ agg, int nE) {
    int e = blockIdx.x * 8 + (threadIdx.x >> 5);
    if (e >= nE) return;
    int lane = threadIdx.x & 31;
    int u  = uidx[e];
    int it = NUSERS + iidx[e];
    int t  = tseq[e];
    float nm = nrm[e];
    float4 ef = ((const float4*)(tt + (size_t)t  * H))[lane];
    float4 xu = ((const float4*)(x  + (size_t)u  * H))[lane];
    float4 xi = ((const float4*)(x  + (size_t)it * H))[lane];
    float* ai = agg + (size_t)it * H + lane * 4;
    float* au = agg + (size_t)u  * H + lane * 4;
    atomAddF(ai + 0, (xu.x + ef.x) * nm);
    atomAddF(ai + 1, (xu.y + ef.y) * nm);
    atomAddF(ai + 2, (xu.z + ef.z) * nm);
    atomAddF(ai + 3, (xu.w + ef.w) * nm);
    atomAddF(au + 0, (xi.x + ef.x) * nm);
    atomAddF(au + 1, (xi.y + ef.y) * nm);
    atomAddF(au + 2, (xi.z + ef.z) * nm);
    atomAddF(au + 3, (xi.w + ef.w) * nm);
}

// ---------------- 6) GEMM + bias + leaky_relu, h written IN PLACE over agg ----------------
// Block: 8 waves; 16 rows x 128 cols per block; wave w owns col tile [16w,16w+16).
// W_l staged in LDS as K-pair packed float2: Wp[(k>>1)*128 + c] = {W[k][c], W[k+1][c]}
// -> B operand is one ds_load_b64 per v_wmma_f32_16x16x4_f32.
__global__ __launch_bounds__(256) void gemm_kernel(float* agg,                      // read [N,H], write h in place
                                                   const float* __restrict__ Wl,   // [H,H]
                                                   const float* __restrict__ bl) { // [H]
    __shared__ float Wp[H * H];             // exactly 64 KB
    int tid = threadIdx.x;
    // cooperative load + K-pair pack
    for (int idx = tid; idx < H * H; idx += 256) {
        int k = idx >> 7, c = idx & 127;
        Wp[(((k >> 1) * H) + c) * 2 + (k & 1)] = Wl[idx];
    }
    __syncthreads();

    int wave  = tid >> 5;
    int lane  = tid & 31;
    int l16   = lane & 15;                  // A: row-in-tile, B: col-in-tile
    int khalf = (lane >> 4) << 1;           // 0 for lanes 0-15, 2 for lanes 16-31
    int colg  = wave * 16 + l16;
    int row0  = blockIdx.x * 16;

    const float* aptr = agg + (size_t)(row0 + l16) * H + khalf;
    const float* bptr = Wp + (size_t)(khalf >> 1) * 2 * H + 2 * colg; // slot for (k0=0, khalf)

    v8f acc = {};
    #pragma unroll
    for (int k0 = 0; k0 < H; k0 += 4) {
        float2 av = *(const float2*)(aptr + k0);
        float2 bv = *(const float2*)(bptr + k0 * H / 2);  // slot (k0/4 + khalf/2)*128 + colg
        v2f a; a.x = av.x; a.y = av.y;
        v2f b; b.x = bv.x; b.y = bv.y;
        acc = __builtin_amdgcn_wmma_f32_16x16x4_f32(false, a, false, b,
                                                    (short)0, acc, false, false);
    }

    float bias = bl[colg];
    __syncthreads();   // all reads of this block's 16 agg rows complete before in-place writes

    int mbase = (lane >> 4) << 3;           // D: VGPR v holds rows v (lanes 0-15) / v+8 (lanes 16-31)
    float* hbase = agg + (size_t)(row0 + mbase) * H + colg;
    #pragma unroll
    for (int v = 0; v < 8; ++v) {
        float hv = acc[v] + bias;
        hv = (hv > 0.0f) ? hv : SLOPE * hv;
        hbase[(size_t)v * H] = hv;
    }
}

// ---------------- 7) row l2-normalize h -> out[:, (layer+1)*128 : +128] ----------------
__global__ __launch_bounds__(256) void l2norm_kernel(const float* __restrict__ h,
                                                     float* __restrict__ out, int layer) {
    int row = blockIdx.x * 8 + (threadIdx.x >> 5);
    if (row >= NNODES) return;
    int lane = threadIdx.x & 31;
    float4 v = ((const float4*)(h + (size_t)row * H))[lane];
    float s = v.x * v.x + v.y * v.y + v.z * v.z + v.w * v.w;
    #pragma unroll
    for (int off = 16; off >= 1; off >>= 1) s += __shfl_xor(s, off, 32);
    float scale = 1.0f / fmaxf(sqrtf(s), 1e-12f);
    float4 o; o.x = v.x * scale; o.y = v.y * scale; o.z = v.z * scale; o.w = v.w * scale;
    ((float4*)(out + (size_t)row * OUTW + (size_t)(layer + 1) * H))[lane] = o;
}

// ---------------- host launcher ----------------
extern "C" void kernel_launch(void* const* d_in, const int* in_sizes, int n_in,
                              void* d_out, int out_size, void* d_ws, size_t ws_size,
                              hipStream_t stream) {
    (void)in_sizes; (void)n_in; (void)out_size; (void)ws_size;
    const float* user_embd  = (const float*)d_in[0];
    const float* item_embd  = (const float*)d_in[1];
    const float* time_table = (const float*)d_in[2];
    const float* Wmat       = (const float*)d_in[3];   // [3,128,128]
    const float* bvec       = (const float*)d_in[4];   // [3,128]
    const int*   user_idx   = (const int*)d_in[5];
    const int*   item_idx   = (const int*)d_in[6];
    const int*   time_seq   = (const int*)d_in[7];
    float* out = (float*)d_out;

    const size_t XB = (size_t)NNODES * H * sizeof(float);   // 76,800,000 B
    char* w = (char*)d_ws;
    float* buf0 = (float*)(w);
    float* buf1 = (float*)(w + XB);
    float* dinv = (float*)(w + 2 * XB);                     // 600,000 B
    float* enrm = (float*)(w + 2 * XB + 600064);            // 4,000,000 B

    // degree -> dinv -> per-edge norm (reused for all layers / both directions)
    hipMemsetAsync(dinv, 0, (size_t)NNODES * sizeof(float), stream);
    deg_kernel  <<<(NEDGES + 255) / 256, 256, 0, stream>>>(user_idx, item_idx, dinv, NEDGES);
    dinv_kernel <<<(NNODES + 255) / 256, 256, 0, stream>>>(dinv, NNODES);
    enorm_kernel<<<(NEDGES + 255) / 256, 256, 0, stream>>>(user_idx, item_idx, dinv, enrm, NEDGES);

    // x0 = embeddings; out[:,0:128] = embeddings
    init_kernel<<<(NNODES * 32 + 255) / 256, 256, 0, stream>>>(user_embd, item_embd, buf0, out);

    float* xcur = buf0;
    float* agg  = buf1;
    for (int l = 0; l < NLAYERS; ++l) {
        hipMemsetAsync(agg, 0, XB, stream);
        edge_kernel<<<(NEDGES + 7) / 8, 256, 0, stream>>>(xcur, time_table, enrm,
                                                          user_idx, item_idx, time_seq,
                                                          agg, NEDGES);
        gemm_kernel<<<NNODES / 16, 256, 0, stream>>>(agg,
                                                     Wmat + (size_t)l * H * H,
                                                     bvec + (size_t)l * H);
        l2norm_kernel<<<NNODES / 8, 256, 0, stream>>>(agg, out, l);
        // next layer consumes the un-normalized h (now in agg, written in place)
        float* t = xcur; xcur = agg; agg = t;
    }
}